// Dense_GCN_11793980195110
// MI455X (gfx1250) — compile-verified
//
#include <hip/hip_runtime.h>

typedef __attribute__((ext_vector_type(2))) float v2f;
typedef __attribute__((ext_vector_type(8))) float v8f;

// ---------------------------------------------------------------------------
// WMMA fp32 GEMM: C[M,Nout] = A[M,K] @ B[K,Nout] (B row-major), optional
// fused bias+ReLU epilogue. One wave32 computes one 16x16 output tile using
// V_WMMA_F32_16X16X4_F32 (fp32 in / fp32 accum -> matches fp32 reference).
//
// Per-lane operand layout (CDNA5 ISA 7.12.2, wave32):
//   A 16x4  : lanes 0-15  -> row m0+l, K = {k0,k0+1}
//             lanes 16-31 -> row m0+l, K = {k0+2,k0+3}
//   B 4x16  : VGPR r, lanes 0-15  -> K=k0+r,   N=n0+l
//             VGPR r, lanes 16-31 -> K=k0+2+r, N=n0+l
//   C/D     : VGPR r, lanes 0-15 -> M=m0+r, lanes 16-31 -> M=m0+r+8, N=n0+l
// ---------------------------------------------------------------------------
__global__ void gemm16_wmma(const float* __restrict__ A, int lda, int acol,
                            const float* __restrict__ B,
                            const float* __restrict__ bias,
                            float* __restrict__ C, int ldc,
                            int M, int K, int Nout, int fuse) {
  const int lane = threadIdx.x & 31;
  const int wave = threadIdx.x >> 5;
  const int tile = blockIdx.x * (blockDim.x >> 5) + wave;
  const int ntn  = Nout >> 4;               // N tiles
  const int mt   = tile / ntn;
  const int nt   = tile - mt * ntn;
  const int m0   = mt << 4;
  if (m0 >= M) return;                      // wave-uniform guard
  const int n0   = nt << 4;
  const int half = lane >> 4;               // 0: lanes 0-15, 1: lanes 16-31
  const int l16  = lane & 15;

  const int  arow = m0 + l16;
  const bool aok  = arow < M;
  const float* Ap = A + (size_t)(aok ? arow : 0) * lda + acol + 2 * half;
  const float* Bp = B + (size_t)(2 * half) * Nout + n0 + l16;

  v8f acc = {0.f, 0.f, 0.f, 0.f, 0.f, 0.f, 0.f, 0.f};
  for (int k0 = 0; k0 < K; k0 += 4) {
    v2f a = *reinterpret_cast<const v2f*>(Ap + k0);   // A[row][k..k+1] (8B aligned)
    if (!aok) { a.x = 0.f; a.y = 0.f; }               // branchless zero-fill
    v2f b;
    b.x = Bp[(size_t)k0 * Nout];
    b.y = Bp[(size_t)(k0 + 1) * Nout];
    // (neg_a, A, neg_b, B, c_mod, C, reuse_a, reuse_b)
    acc = __builtin_amdgcn_wmma_f32_16x16x4_f32(false, a, false, b,
                                                (short)0, acc, false, false);
  }

  const int   col = n0 + l16;
  const float bv  = fuse ? bias[col] : 0.f;
#pragma unroll
  for (int r = 0; r < 8; ++r) {
    const int row = m0 + r + 8 * half;
    if (row < M) {
      float v = acc[r];
      if (fuse) v = fmaxf(v + bv, 0.f);
      C[(size_t)row * ldc + col] = v;
    }
  }
}

// ---------------------------------------------------------------------------
// Degree / normalization kernels:  deg[v] = 1 + in_degree(v), dinv = rsqrt(deg)
// ---------------------------------------------------------------------------
__global__ void deg_init(float* __restrict__ deg, int n) {
  int i = blockIdx.x * blockDim.x + threadIdx.x;
  if (i < n) deg[i] = 1.0f;                 // self-loop contribution
}

__global__ void deg_edges(const int* __restrict__ dst, float* __restrict__ deg, int E) {
  int e = blockIdx.x * blockDim.x + threadIdx.x;
  if (e < E) atomicAdd(&deg[dst[e]], 1.0f); // global_atomic_add_f32 @ L2
}

__global__ void deg_rsqrt(float* __restrict__ d, int n) {
  int i = blockIdx.x * blockDim.x + threadIdx.x;
  if (i < n) d[i] = rsqrtf(d[i]);           // deg >= 1 always
}

// ---------------------------------------------------------------------------
// agg[v,:] = dinv[v]^2 * h[v,:]   (the appended self-loop edge of the ref)
// ---------------------------------------------------------------------------
__global__ void agg_init(const float* __restrict__ h, const float* __restrict__ dinv,
                         float* __restrict__ agg, long long total, int dshift) {
  long long t = (long long)blockIdx.x * blockDim.x + threadIdx.x;
  if (t >= total) return;
  const int v = (int)(t >> dshift);
  const float w = dinv[v];
  agg[t] = w * w * h[t];
}

// ---------------------------------------------------------------------------
// Edge scatter: agg[dst,:] += dinv[src]*dinv[dst] * h[src,:]
// One thread handles (edge, 4 features): float4 gather + 4 f32 atomics.
// ---------------------------------------------------------------------------
__global__ void agg_edges(const int* __restrict__ src, const int* __restrict__ dst,
                          const float* __restrict__ h, const float* __restrict__ dinv,
                          float* __restrict__ agg, int E, int dshift) {
  const int q = dshift - 2;                 // log2(feature-quads per edge)
  long long t = (long long)blockIdx.x * blockDim.x + threadIdx.x;
  const long long total = (long long)E << q;
  if (t >= total) return;
  const int e = (int)(t >> q);
  const int g = (int)(t & ((1 << q) - 1));
  const int s = src[e], d = dst[e];
  const float nrm = dinv[s] * dinv[d];
  const float4 hv = *reinterpret_cast<const float4*>(h + (((size_t)s << dshift) + (g << 2)));
  float* ap = agg + (((size_t)d << dshift) + (g << 2));
  atomicAdd(ap + 0, nrm * hv.x);
  atomicAdd(ap + 1, nrm * hv.y);
  atomicAdd(ap + 2, nrm * hv.z);
  atomicAdd(ap + 3, nrm * hv.w);
}

// ---------------------------------------------------------------------------
// cat[v, colofs + j] = relu(agg[v,j] + bias[j])
// ---------------------------------------------------------------------------
__global__ void finalize_relu(const float* __restrict__ agg, const float* __restrict__ bias,
                              float* __restrict__ cat, long long total, int dshift,
                              int ldc, int colofs) {
  long long t = (long long)blockIdx.x * blockDim.x + threadIdx.x;
  if (t >= total) return;
  const int v = (int)(t >> dshift);
  const int j = (int)(t & ((1 << dshift) - 1));
  cat[(size_t)v * ldc + colofs + j] = fmaxf(agg[t] + bias[j], 0.0f);
}

// ---------------------------------------------------------------------------
static inline unsigned cdiv(long long a, int b) { return (unsigned)((a + b - 1) / b); }

extern "C" void kernel_launch(void* const* d_in, const int* in_sizes, int n_in,
                              void* d_out, int out_size, void* d_ws, size_t ws_size,
                              hipStream_t stream) {
  const int*   edges = (const int*)d_in[0];   // [2,E] int32
  const float* X     = (const float*)d_in[1]; // [N,128]
  const float* W1    = (const float*)d_in[2];
  const float* b1    = (const float*)d_in[3];
  const float* W2    = (const float*)d_in[4];
  const float* b2    = (const float*)d_in[5];
  const float* W3    = (const float*)d_in[6];
  const float* b3    = (const float*)d_in[7];
  const float* Wfc   = (const float*)d_in[8];
  const float* bfc   = (const float*)d_in[9];

  const int E = in_sizes[0] / 2;
  const int n = in_sizes[1] / 128;
  const int* src = edges;
  const int* dst = edges + E;

  // Workspace carve-up (floats): dinv[n] | h[n*64] | agg[n*64] | cat[n*112]
  float* dinv = (float*)d_ws;
  float* h    = dinv + (((size_t)n + 63) & ~(size_t)63);
  float* agg  = h    + (size_t)n * 64;
  float* cat  = agg  + (size_t)n * 64;
  float* out  = (float*)d_out;                // [N,16]

  const int TB = 256;

  // --- degree normalization ------------------------------------------------
  deg_init <<<cdiv(n, TB), TB, 0, stream>>>(dinv, n);
  deg_edges<<<cdiv(E, TB), TB, 0, stream>>>(dst, dinv, E);
  deg_rsqrt<<<cdiv(n, TB), TB, 0, stream>>>(dinv, n);

  const int mtiles = (n + 15) >> 4;           // 6250

  // --- per-layer pipeline --------------------------------------------------
  struct Layer { const float* A; int lda, acol; const float* W; const float* b;
                 int K, dout, dshift, colofs; };
  const Layer layers[3] = {
    { X,   128,  0, W1, b1, 128, 64, 6,  0 },   // -> cat[:,  0: 64]
    { cat, 112,  0, W2, b2,  64, 32, 5, 64 },   // -> cat[:, 64: 96]
    { cat, 112, 64, W3, b3,  32, 16, 4, 96 },   // -> cat[:, 96:112]
  };

  for (int l = 0; l < 3; ++l) {
    const Layer& L = layers[l];
    const long long tiles = (long long)mtiles * (L.dout >> 4);
    // h = A @ W   (no epilogue)
    gemm16_wmma<<<cdiv(tiles * 32, 128), 128, 0, stream>>>(
        L.A, L.lda, L.acol, L.W, L.b, h, L.dout, n, L.K, L.dout, 0);
    const long long tot = (long long)n << L.dshift;
    agg_init<<<cdiv(tot, TB), TB, 0, stream>>>(h, dinv, agg, tot, L.dshift);
    const long long etot = (long long)E << (L.dshift - 2);
    agg_edges<<<cdiv(etot, TB), TB, 0, stream>>>(src, dst, h, dinv, agg, E, L.dshift);
    finalize_relu<<<cdiv(tot, TB), TB, 0, stream>>>(agg, L.b, cat, tot, L.dshift,
                                                    112, L.colofs);
  }

  // --- FC head: out = relu(cat @ Wfc + bfc), fused epilogue ---------------
  const long long fct = (long long)mtiles * 1;  // Nout=16 -> 1 N-tile
  gemm16_wmma<<<cdiv(fct * 32, 128), 128, 0, stream>>>(
      cat, 112, 0, Wfc, bfc, out, 16, n, 112, 16, 1);
}